// SO3GridGen_42202348650904
// MI455X (gfx1250) — compile-verified
//
#include <hip/hip_runtime.h>
#include <math.h>

// Problem constants (match reference)
#define FXc 240.0f
#define FYc 240.0f
#define U0c 120.0f
#define V0c 160.0f
#define Hc  240
#define Wc  320
#define Cc  64
#define Nb  16

typedef __attribute__((ext_vector_type(2))) float v2f;
typedef __attribute__((ext_vector_type(8))) float v8f;

// One wave owns 32 consecutive pixels. Phase 1: each lane computes its own
// pixel's warp coordinates/weights (fully vectorized). Phase 2: uniform loop
// over the 32 pixels; per pixel the wave gathers the 4 bilinear taps for all
// 64 channels and blends them with exact-f32 V_WMMA_F32_16X16X4_F32.
__global__ __launch_bounds__(256) void so3_warp_wmma(
    const float* __restrict__ so3,
    const float* __restrict__ img,
    float* __restrict__ out)
{
  const int lane = threadIdx.x & 31;
  const int hi   = lane >> 4;     // 0: lanes 0-15 (K=0,1) ; 1: lanes 16-31 (K=2,3)
  const int n    = lane & 15;     // channel-within-group / matrix column
  const long gtid = (long)blockIdx.x * blockDim.x + threadIdx.x;
  const int wave  = (int)(gtid >> 5);
  const int p     = wave * 32 + lane;          // this lane's pixel (flat b*H*W + h*W + w)

  // ---------------- phase 1: per-lane coordinate math ----------------
  const int wx_i = p % Wc;
  const int hw   = p / Wc;
  const int h_i  = hw % Hc;
  const int b_i  = hw / Hc;

  const float a  = so3[b_i * 3 + 0];
  const float bb = so3[b_i * 3 + 1];
  const float c  = so3[b_i * 3 + 2];
  const float t2    = a * a + bb * bb + c * c;
  const float theta = sqrtf(t2 + 1e-12f);
  const float sA = sinf(theta) / theta;
  const float sB = (1.0f - cosf(theta)) / (t2 + 1e-12f);
  // R = I + sA*K + sB*K^2 ;  K = [[0,-c,b],[c,0,-a],[-b,a,0]]
  const float R00 = 1.0f + sB * (-(bb * bb + c * c));
  const float R01 = -sA * c + sB * (a * bb);
  const float R02 =  sA * bb + sB * (a * c);
  const float R10 =  sA * c + sB * (a * bb);
  const float R11 = 1.0f + sB * (-(a * a + c * c));
  const float R12 = -sA * a + sB * (bb * c);
  const float R20 = -sA * bb + sB * (a * c);
  const float R21 =  sA * a + sB * (bb * c);
  const float R22 = 1.0f + sB * (-(a * a + bb * bb));

  const float X = ((float)wx_i - U0c) / FXc;
  const float Y = ((float)h_i  - V0c) / FYc;
  const float Pr0 = R00 * X + R01 * Y + R02;
  const float Pr1 = R10 * X + R11 * Y + R12;
  const float Z   = R20 * X + R21 * Y + R22;
  const float Zs  = (fabsf(Z) < 1e-6f) ? 1e-6f : Z;
  const float xc  = FXc * Pr0 / Zs + U0c;
  const float yc  = FYc * Pr1 / Zs + V0c;
  // reference's normalize/denormalize round-trip (algebraically identity)
  const float xn  = 2.0f * xc / (float)(Wc - 1) - 1.0f;
  const float yn  = 2.0f * yc / (float)(Hc - 1) - 1.0f;
  const float pxf = (xn + 1.0f) * (float)(Wc - 1) * 0.5f;
  const float pyf = (yn + 1.0f) * (float)(Hc - 1) * 0.5f;
  const float x0f = floorf(pxf);
  const float y0f = floorf(pyf);
  const float fx  = pxf - x0f;
  const float fy  = pyf - y0f;
  const int x0i = (int)fminf(fmaxf(x0f, 0.0f), (float)(Wc - 1));
  const int y0i = (int)fminf(fmaxf(y0f, 0.0f), (float)(Hc - 1));
  const int dx  = (x0i + 1 <= Wc - 1) ? 1 : 0;   // x1i = x0i + dx (clamped)
  const int dy  = (y0i + 1 <= Hc - 1) ? 1 : 0;   // y1i = y0i + dy (clamped)

  const int base00 = ((b_i * Hc + y0i) * Wc + x0i) * Cc;  // element offset of tap (y0,x0)
  const int dxC    = dx * Cc;                              // step to x1 column
  const int dyWC   = dy * Wc * Cc;                         // step to y1 row
  const int ooff   = p * Cc;                               // output element offset

  const int fx_bits = (int)__builtin_bit_cast(unsigned, fx);
  const int fy_bits = (int)__builtin_bit_cast(unsigned, fy);

  // ---------------- phase 2: uniform per-pixel loop ----------------
  for (int q = 0; q < 32; ++q) {
    const float s_fx = __builtin_bit_cast(float, (unsigned)__builtin_amdgcn_readlane(fx_bits, q));
    const float s_fy = __builtin_bit_cast(float, (unsigned)__builtin_amdgcn_readlane(fy_bits, q));
    const int   s_b00 = __builtin_amdgcn_readlane(base00, q);
    const int   s_dxC = __builtin_amdgcn_readlane(dxC,   q);
    const int   s_dyW = __builtin_amdgcn_readlane(dyWC,  q);
    const int   s_out = __builtin_amdgcn_readlane(ooff,  q);

    // A (16x4 f32): every row = this pixel's 4 bilinear weights.
    // lanes 0-15:  {K0,K1} = {w00,w01} ; lanes 16-31: {K2,K3} = {w10,w11}
    const float trow = hi ? s_fy : (1.0f - s_fy);
    v2f A;
    A.x = trow * (1.0f - s_fx);
    A.y = trow * s_fx;

    // B (4x16 f32): VGPR0 = {K0 | K2}, VGPR1 = {K1 | K3}; column n = channel.
    // K0=(y0,x0) K1=(y0,x1) K2=(y1,x0) K3=(y1,x1)
    const float* p0 = img + (s_b00 + (hi ? s_dyW : 0) + n);   // x0 column of my row
    const float* p1 = p0 + s_dxC;                             // x1 column of my row
    v2f B0, B1, B2, B3;
    B0.x = p0[0];   B0.y = p1[0];     // channels  0..15
    B1.x = p0[16];  B1.y = p1[16];    // channels 16..31
    B2.x = p0[32];  B2.y = p1[32];    // channels 32..47
    B3.x = p0[48];  B3.y = p1[48];    // channels 48..63

    v8f zero = {};
    v8f d0 = __builtin_amdgcn_wmma_f32_16x16x4_f32(false, A, false, B0, (short)0, zero, false, false);
    v8f d1 = __builtin_amdgcn_wmma_f32_16x16x4_f32(false, A, false, B1, (short)0, zero, false, false);
    v8f d2 = __builtin_amdgcn_wmma_f32_16x16x4_f32(false, A, false, B2, (short)0, zero, false, false);
    v8f d3 = __builtin_amdgcn_wmma_f32_16x16x4_f32(false, A, false, B3, (short)0, zero, false, false);

    // D row 0 lives in d[0], lanes 0-15 (N = lane = channel-in-group).
    if (lane < 16) {
      float* po = out + (s_out + lane);
      po[0]  = d0[0];
      po[16] = d1[0];
      po[32] = d2[0];
      po[48] = d3[0];
    }
  }
}

extern "C" void kernel_launch(void* const* d_in, const int* in_sizes, int n_in,
                              void* d_out, int out_size, void* d_ws, size_t ws_size,
                              hipStream_t stream) {
  (void)in_sizes; (void)n_in; (void)out_size; (void)d_ws; (void)ws_size;
  const float* so3 = (const float*)d_in[0];
  const float* img = (const float*)d_in[1];
  float* out = (float*)d_out;

  const int pixels  = Nb * Hc * Wc;          // 1,228,800 (divisible by 32*8)
  const int threads = 256;                   // 8 waves
  const int blocks  = pixels / (32 * (threads / 32));  // 4800
  so3_warp_wmma<<<blocks, threads, 0, stream>>>(so3, img, out);
}